// SpecialSpmm_42949673125
// MI455X (gfx1250) — compile-verified
//
#include <hip/hip_runtime.h>
#include <hip/hip_bf16.h>
#include <stdint.h>

// COO SpMM: out[N,128] = A @ b, A sparse COO (rows, cols, values), E edges.
// Memory-bound scatter workload: b (25.6MB) and out (25.6MB) are L2-resident
// (192MB L2). Dominant cost = ~205M f32 atomics into L2. WMMA is useless here
// (0.4 GFLOP total, AI ~0.25 FLOP/B); the CDNA5 win is a 4-deep async
// global->LDS gather pipeline (ASYNCcnt) + NT hints on the streamed edge data.

#define WAVES_PER_BLOCK 8
#define BLOCK_THREADS (WAVES_PER_BLOCK * 32)
#define PIPE 4           // async pipeline depth (circular LDS buffers)

__global__ void __launch_bounds__(256) zero_f32_kernel(float* __restrict__ out, int n) {
    int i = blockIdx.x * blockDim.x + threadIdx.x;
    int i4 = i * 4;
    if (i4 + 3 < n) {
        *reinterpret_cast<float4*>(out + i4) = make_float4(0.f, 0.f, 0.f, 0.f);
    } else {
        for (int k = i4; k < n; ++k) out[k] = 0.f;
    }
}

__global__ void __launch_bounds__(BLOCK_THREADS) spmm_coo_kernel(
    const int* __restrict__ rows,
    const int* __restrict__ cols,
    const float* __restrict__ vals,
    const float* __restrict__ b,    // [N,128]
    float* __restrict__ out,        // [N,128]
    int E) {

    // 4-deep circular per-wave staging of 512B rows of b.
    __shared__ float lds[WAVES_PER_BLOCK][PIPE][128];

    const int lane = threadIdx.x & 31;
    const int w    = threadIdx.x >> 5;
    const int wave_id = (blockIdx.x * blockDim.x + threadIdx.x) >> 5;
    const int n_waves = (gridDim.x * blockDim.x) >> 5;

    // LDS byte offset of this lane's 16B slice in buffer 0 (flat->LDS: low 32
    // bits of a shared pointer are the workgroup-relative LDS offset, which is
    // exactly what the async-to-LDS VDST operand expects). Buffer k is at
    // +k*512 bytes.
    const uint32_t lds_base =
        (uint32_t)(uintptr_t)(void*)&lds[w][0][lane * 4];

    for (int base = wave_id * 32; base < E; base += n_waves * 32) {
        const int nval = min(32, E - base);

        // Coalesced, non-temporal load of this tile's edge metadata
        // (streamed once -> keep it out of L2; b/out want the L2).
        int   r = 0, c = 0;
        float v = 0.f;
        if (lane < nval) {
            const int e = base + lane;
            r = __builtin_nontemporal_load(rows + e);
            c = __builtin_nontemporal_load(cols + e);
            v = __builtin_nontemporal_load(vals + e);
        }

        if (nval == 32) {
            // ---- full tile: 4-deep async gather pipeline (EXEC all-1s) ----
            // Prologue: issue gathers for edges 0..2 into buffers 0..2.
            #pragma unroll
            for (int p = 0; p < PIPE - 1; ++p) {
                const int cp = __builtin_amdgcn_readlane(c, p);
                const float* src = b + (size_t)cp * 128 + lane * 4;
                const uint32_t dst = lds_base + (uint32_t)p * 512u;
                asm volatile("global_load_async_to_lds_b128 %0, %1, off"
                             :: "v"(dst), "v"(src) : "memory");
            }
            #pragma unroll
            for (int i = 0; i < 32; ++i) {
                // Issue gather for edge i+3 into buffer (i+3)%4.
                if (i + PIPE - 1 < 32) {
                    const int cn = __builtin_amdgcn_readlane(c, i + PIPE - 1);
                    const float* srcn = b + (size_t)cn * 128 + lane * 4;
                    const uint32_t dsti =
                        lds_base + (uint32_t)((i + PIPE - 1) & (PIPE - 1)) * 512u;
                    asm volatile("global_load_async_to_lds_b128 %0, %1, off"
                                 :: "v"(dsti), "v"(srcn) : "memory");
                }
                // Async loads complete in order: counter <= rem outstanding
                // younger loads  =>  buffer i is resident.
                const int rem = 31 - i;
                if (rem >= 3)      asm volatile("s_wait_asynccnt 3" ::: "memory");
                else if (rem == 2) asm volatile("s_wait_asynccnt 2" ::: "memory");
                else if (rem == 1) asm volatile("s_wait_asynccnt 1" ::: "memory");
                else               asm volatile("s_wait_asynccnt 0" ::: "memory");

                const int   ri = __builtin_amdgcn_readlane(r, i);
                const float vi = __uint_as_float(
                    __builtin_amdgcn_readlane(__float_as_uint(v), i));

                const float4 x = *reinterpret_cast<const float4*>(
                    &lds[w][i & (PIPE - 1)][lane * 4]);

                float* dst = out + (size_t)ri * 128 + lane * 4;
                atomicAdd(dst + 0, vi * x.x);
                atomicAdd(dst + 1, vi * x.y);
                atomicAdd(dst + 2, vi * x.z);
                atomicAdd(dst + 3, vi * x.w);
            }
        } else {
            // ---- tail tile: plain gather path ----
            for (int i = 0; i < nval; ++i) {
                const int   ri = __builtin_amdgcn_readlane(r, i);
                const int   ci = __builtin_amdgcn_readlane(c, i);
                const float vi = __uint_as_float(
                    __builtin_amdgcn_readlane(__float_as_uint(v), i));

                const float4 x = *reinterpret_cast<const float4*>(
                    b + (size_t)ci * 128 + lane * 4);

                float* dst = out + (size_t)ri * 128 + lane * 4;
                atomicAdd(dst + 0, vi * x.x);
                atomicAdd(dst + 1, vi * x.y);
                atomicAdd(dst + 2, vi * x.z);
                atomicAdd(dst + 3, vi * x.w);
            }
        }
    }
}

extern "C" void kernel_launch(void* const* d_in, const int* in_sizes, int n_in,
                              void* d_out, int out_size, void* d_ws, size_t ws_size,
                              hipStream_t stream) {
    (void)n_in; (void)d_ws; (void)ws_size;

    const int E = in_sizes[1];                    // values: [E]
    const int* indices = (const int*)d_in[0];     // [2, E] flattened
    const int* rows = indices;
    const int* cols = indices + E;
    const float* vals = (const float*)d_in[1];
    const float* b    = (const float*)d_in[3];    // [N, 128]
    float* out        = (float*)d_out;            // [N, 128]

    // 1) zero the output (harness poisons d_out; atomics accumulate into it)
    {
        int n4 = (out_size + 3) / 4;
        int blocks = (n4 + 255) / 256;
        if (blocks < 1) blocks = 1;
        zero_f32_kernel<<<blocks, 256, 0, stream>>>(out, out_size);
    }

    // 2) edge kernel: one wave32 per 32-edge tile, grid-stride
    {
        int tiles  = (E + 31) / 32;
        int blocks = (tiles + WAVES_PER_BLOCK - 1) / WAVES_PER_BLOCK;
        if (blocks < 1) blocks = 1;
        if (blocks > 16384) blocks = 16384;
        spmm_coo_kernel<<<blocks, BLOCK_THREADS, 0, stream>>>(
            rows, cols, vals, b, out, E);
    }
}